// DE_NN_21466246545864
// MI455X (gfx1250) — compile-verified
//
#include <hip/hip_runtime.h>

// DE population update: pure streaming elementwise op, HBM-bandwidth bound.
// ~92.4M fp32 params, ~2.2GB moved/call -> ~95us floor at 23.3 TB/s.
// MI455X path: b128 global ops, NT temporal hints (TH=NT) for single-use
// streams (ri, output) so the 192MB L2 keeps x resident for best/a0/a1 row
// gathers (w1 row-set = 184.5MB fits), uniform early-exit for rejected rows,
// 64B per thread for deep load clauses / MLP. No matrix FLOPs exist in this
// op, so WMMA/TDM are structurally inapplicable.

typedef __attribute__((ext_vector_type(4))) float vf4;

#define CR_CONST 0.9f
#define F_CONST  0.8f
#define NP_POP   44
#define ITEMS    4      // float4s per thread
#define BLK      256    // threads per block (8 wave32)

__global__ __launch_bounds__(BLK) void de_crossover_kernel(
    const float* __restrict__ X,   // [NP, S] parameter tensor
    const float* __restrict__ RI,  // [NP, S] uniform randoms
    float*       __restrict__ OUT, // [NP, S] updated tensor
    int S4,                        // S/4 (float4 count per population row)
    const float* __restrict__ fx,
    const float* __restrict__ fy,
    const int*   __restrict__ best_p,
    const int*   __restrict__ a0,
    const int*   __restrict__ a1,
    const int*   __restrict__ Rs,
    int layer)
{
    const int p    = blockIdx.y;
    const int base = blockIdx.x * (BLK * ITEMS) + threadIdx.x;

    const vf4* __restrict__ Xv  = (const vf4*)X;
    const vf4* __restrict__ RIv = (const vf4*)RI;
    vf4*       __restrict__ Ov  = (vf4*)OUT;

    const long long rowX = (long long)p * S4;

    // Uniform per-row scalars (s_load path).
    const bool accept = (fy[p] <= fx[p]);

    if (!accept) {
        // Rejected row: copy-through. 1 read + 1 NT write per element.
        #pragma unroll
        for (int k = 0; k < ITEMS; ++k) {
            const int r4 = base + k * BLK;
            if (r4 < S4) {
                const vf4 x = Xv[rowX + r4];
                __builtin_nontemporal_store(x, &Ov[rowX + r4]);
            }
        }
        return;
    }

    const long long rowB = (long long)best_p[0] * S4;
    const long long row0 = (long long)a0[p]     * S4;
    const long long row1 = (long long)a1[p]     * S4;
    const bool force = (Rs[p] == layer);

    // Phase 1: issue all loads (up to 20 b128 in flight) before any compute.
    vf4 x[ITEMS], xb[ITEMS], x0[ITEMS], x1[ITEMS], ri[ITEMS];
    bool v[ITEMS];
    #pragma unroll
    for (int k = 0; k < ITEMS; ++k) {
        const int r4 = base + k * BLK;
        v[k] = (r4 < S4);
        if (v[k]) {
            x[k]  = Xv[rowX + r4];                            // RT: L2-resident
            xb[k] = Xv[rowB + r4];                            // RT: broadcast row
            x0[k] = Xv[row0 + r4];                            // RT: gather rows
            x1[k] = Xv[row1 + r4];
            ri[k] = __builtin_nontemporal_load(&RIv[rowX + r4]); // NT: single-use
        }
    }

    // Phase 2: compute + NT store.
    #pragma unroll
    for (int k = 0; k < ITEMS; ++k) {
        if (!v[k]) continue;
        const int r4 = base + k * BLK;
        // mutant = x + F*(xb - x) + F*(x0 - x1)
        const vf4 m = x[k] + F_CONST * (xb[k] - x[k]) + F_CONST * (x0[k] - x1[k]);
        vf4 o;
        o.x = (force || (ri[k].x < CR_CONST)) ? m.x : x[k].x;
        o.y = (force || (ri[k].y < CR_CONST)) ? m.y : x[k].y;
        o.z = (force || (ri[k].z < CR_CONST)) ? m.z : x[k].z;
        o.w = (force || (ri[k].w < CR_CONST)) ? m.w : x[k].w;
        __builtin_nontemporal_store(o, &Ov[rowX + r4]);
    }
}

// new_fx = where(fy<=fx, fy, fx); min_f = min(new_fx); best = argmin (first occurrence)
__global__ __launch_bounds__(64) void de_select_kernel(
    const float* __restrict__ fx, const float* __restrict__ fy,
    float* __restrict__ out_fx, float* __restrict__ out_minf,
    float* __restrict__ out_best)
{
    __shared__ float nf[NP_POP];
    const int t = threadIdx.x;
    if (t < NP_POP) {
        const float v = (fy[t] <= fx[t]) ? fy[t] : fx[t];
        nf[t] = v;
        out_fx[t] = v;
    }
    __syncthreads();
    if (t == 0) {
        float mn = nf[0];
        int   bi = 0;
        for (int i = 1; i < NP_POP; ++i) {
            if (nf[i] < mn) { mn = nf[i]; bi = i; }  // strict '<' => first occurrence
        }
        out_minf[0] = mn;
        out_best[0] = (float)bi;
    }
}

extern "C" void kernel_launch(void* const* d_in, const int* in_sizes, int n_in,
                              void* d_out, int out_size, void* d_ws, size_t ws_size,
                              hipStream_t stream) {
    (void)in_sizes; (void)n_in; (void)out_size; (void)d_ws; (void)ws_size;

    // setup_inputs() dict order:
    //  0:w0 1:b0 2:ri_w0 3:ri_b0 4:w1 5:b1 6:ri_w1 7:ri_b1
    //  8:w2 9:b2 10:ri_w2 11:ri_b2 12:fx 13:fy 14:best_model 15:a0 16:a1 17:Rs
    const float* w0    = (const float*)d_in[0];
    const float* b0    = (const float*)d_in[1];
    const float* ri_w0 = (const float*)d_in[2];
    const float* ri_b0 = (const float*)d_in[3];
    const float* w1    = (const float*)d_in[4];
    const float* b1    = (const float*)d_in[5];
    const float* ri_w1 = (const float*)d_in[6];
    const float* ri_b1 = (const float*)d_in[7];
    const float* w2    = (const float*)d_in[8];
    const float* b2    = (const float*)d_in[9];
    const float* ri_w2 = (const float*)d_in[10];
    const float* ri_b2 = (const float*)d_in[11];
    const float* fx    = (const float*)d_in[12];
    const float* fy    = (const float*)d_in[13];
    const int*   best  = (const int*)d_in[14];
    const int*   a0    = (const int*)d_in[15];
    const int*   a1    = (const int*)d_in[16];
    const int*   Rs    = (const int*)d_in[17];

    // Per-row slice sizes (elements)
    const long long S_w0 = 1024LL * 512;   // 524288
    const long long S_b0 = 1024LL;
    const long long S_w1 = 1024LL * 1024;  // 1048576
    const long long S_b1 = 1024LL;
    const long long S_w2 = 512LL * 1024;   // 524288
    const long long S_b2 = 512LL;

    // Output layout: new_w0,new_b0,new_w1,new_b1,new_w2,new_b2,new_fx,min_f,best
    float* out = (float*)d_out;
    float* o_w0 = out;
    float* o_b0 = o_w0 + NP_POP * S_w0;
    float* o_w1 = o_b0 + NP_POP * S_b0;
    float* o_b1 = o_w1 + NP_POP * S_w1;
    float* o_w2 = o_b1 + NP_POP * S_b1;
    float* o_b2 = o_w2 + NP_POP * S_w2;
    float* o_fx   = o_b2 + NP_POP * S_b2;
    float* o_minf = o_fx + NP_POP;
    float* o_best = o_minf + 1;

    auto launch = [&](const float* X, const float* RI, float* O,
                      long long S, int layer) {
        const int S4   = (int)(S / 4);
        const int tile = BLK * ITEMS;
        dim3 grid((S4 + tile - 1) / tile, NP_POP);
        de_crossover_kernel<<<grid, BLK, 0, stream>>>(
            X, RI, O, S4, fx, fy, best, a0, a1, Rs, layer);
    };

    launch(w0, ri_w0, o_w0, S_w0, 0);
    launch(b0, ri_b0, o_b0, S_b0, 0);
    launch(w1, ri_w1, o_w1, S_w1, 1);
    launch(b1, ri_b1, o_b1, S_b1, 1);
    launch(w2, ri_w2, o_w2, S_w2, 2);
    launch(b2, ri_b2, o_b2, S_b2, 2);

    de_select_kernel<<<1, 64, 0, stream>>>(fx, fy, o_fx, o_minf, o_best);
}